// LSTMSetDecoder_76201309766343
// MI455X (gfx1250) — compile-verified
//
#include <hip/hip_runtime.h>
#include <math.h>

// ---------------- problem constants (match reference) ----------------
#define EMB     128
#define HID     512
#define GATES   (4 * HID)     // 2048
#define KAH     (EMB + HID)   // 640: A = [x | h] concatenated along K
#define VTGT    8000
#define NUM_OUT 9
#define BOS     0

// ---------------- WMMA fragment types (fp32 16x16x4) ----------------
typedef __attribute__((ext_vector_type(2))) float v2f;   // A/B frag: 2 VGPRs
typedef __attribute__((ext_vector_type(8))) float v8f;   // C/D frag: 8 VGPRs

// ---------------- tiling ----------------
#define TSM  128    // block C-tile rows (8 waves x 16)
#define TSN  64     // block C-tile cols (per-wave 16x64 -> 4 accumulators)
#define KC   16     // K chunk staged in LDS
#define LSTR 18     // LDS row stride (pad: conflict-free 16-lane reads, 8B aligned)

// =====================================================================
// helpers
// =====================================================================
__device__ __forceinline__ unsigned int float_key(float v) {
    unsigned int b = __float_as_uint(v);
    return (b & 0x80000000u) ? ~b : (b | 0x80000000u);   // monotonic float->u32
}
__device__ __forceinline__ unsigned long long pack_max(float v, int col) {
    // higher value wins; on tie, smaller column wins (first-index argmax)
    return ((unsigned long long)float_key(v) << 32) |
           (unsigned long long)(0xFFFFFFFFu - (unsigned int)col);
}
__device__ __forceinline__ unsigned long long shfl_xor_u64(unsigned long long v, int m) {
    unsigned int lo = (unsigned int)v;
    unsigned int hi = (unsigned int)(v >> 32);
    lo = __shfl_xor(lo, m, 32);
    hi = __shfl_xor(hi, m, 32);
    return ((unsigned long long)hi << 32) | lo;
}
__device__ __forceinline__ float sigmoidf_(float x) { return 1.0f / (1.0f + expf(-x)); }

// =====================================================================
// init: zero AH and cell state, argmax scratch; token = BOS
// =====================================================================
__global__ void init_state_kernel(float* __restrict__ AH, float* __restrict__ C,
                                  int* __restrict__ tok,
                                  unsigned long long* __restrict__ packed, int B) {
    int x = blockIdx.x * blockDim.x + threadIdx.x;
    if (x < B * KAH) AH[x] = 0.0f;
    if (x < B * HID) C[x]  = 0.0f;
    if (x < B) { tok[x] = BOS; packed[x] = 0ULL; }
}

// =====================================================================
// embedding gathers into AH[:, 0:EMB]
// =====================================================================
__global__ void embed_src_kernel(const int* __restrict__ src, const float* __restrict__ emb,
                                 float* __restrict__ AH, int t, int B) {
    int x = blockIdx.x * blockDim.x + threadIdx.x;
    if (x >= B * EMB) return;
    int m = x / EMB, e = x % EMB;
    AH[m * KAH + e] = emb[src[m * 2 + t] * EMB + e];
}
__global__ void embed_tok_kernel(const int* __restrict__ tok, const float* __restrict__ emb,
                                 float* __restrict__ AH, int B) {
    int x = blockIdx.x * blockDim.x + threadIdx.x;
    if (x >= B * EMB) return;
    int m = x / EMB, e = x % EMB;
    AH[m * KAH + e] = emb[tok[m] * EMB + e];
}

// =====================================================================
// gate GEMM: G[B x 2048] = [x|h] @ [Wih|Whh]^T   (K = 640, fp32 WMMA)
// block tile 128x64, 8 waves, wave tile 16x64 (4 accumulators)
// =====================================================================
__global__ __launch_bounds__(256)
void gates_gemm_kernel(const float* __restrict__ AH,   // B x KAH
                       const float* __restrict__ Wih,  // GATES x EMB
                       const float* __restrict__ Whh,  // GATES x HID
                       float* __restrict__ G) {        // B x GATES
    __shared__ float lA[TSM * LSTR];   // 128 x 16 (padded)
    __shared__ float lW[TSN * LSTR];   //  64 x 16 (padded)

    const int tid  = threadIdx.x;
    const int lane = tid & 31;
    const int wv   = tid >> 5;          // 0..7
    const int wm   = wv * 16;           // wave row base within block tile
    const int bm   = blockIdx.x * TSM;
    const int bn   = blockIdx.y * TSN;

    const int lrA = tid >> 1;           // 0..127 : A row (coop load, 8 floats each)
    const int qA  = (tid & 1) * 8;      // 0 or 8
    const int lrW = tid >> 2;           // 0..63  : W row (coop load, 4 floats each)
    const int qW  = (tid & 3) * 4;      // 0,4,8,12

    v8f acc[4] = {v8f{}, v8f{}, v8f{}, v8f{}};

    for (int kb = 0; kb < KAH; kb += KC) {
        // stage A chunk (128 x 16)
        {
            const float* sp = &AH[(bm + lrA) * KAH + kb + qA];
            float4 p0 = ((const float4*)sp)[0];
            float4 p1 = ((const float4*)sp)[1];
            float* d = &lA[lrA * LSTR + qA];
            ((float2*)d)[0]     = make_float2(p0.x, p0.y);
            ((float2*)(d + 2))[0] = make_float2(p0.z, p0.w);
            ((float2*)(d + 4))[0] = make_float2(p1.x, p1.y);
            ((float2*)(d + 6))[0] = make_float2(p1.z, p1.w);
        }
        // stage W chunk (64 x 16): B[k][n] = W[n][k]; chunks never straddle EMB
        {
            const float* wrow; int ko;
            if (kb < EMB) { wrow = &Wih[(bn + lrW) * EMB]; ko = kb + qW; }
            else          { wrow = &Whh[(bn + lrW) * HID]; ko = kb - EMB + qW; }
            const float4 p = *(const float4*)&wrow[ko];
            float* d = &lW[lrW * LSTR + qW];
            ((float2*)d)[0]       = make_float2(p.x, p.y);
            ((float2*)(d + 2))[0] = make_float2(p.z, p.w);
        }
        __syncthreads();

        const int m  = wm + (lane & 15);
        const int nl = lane & 15;
        const int kh = (lane >> 4) * 2;          // K split across lane halves
#pragma unroll
        for (int kk = 0; kk < 4; ++kk) {
            const int k0 = kk * 4 + kh;
            v2f a = *(const v2f*)&lA[m * LSTR + k0];
#pragma unroll
            for (int s = 0; s < 4; ++s) {
                v2f b = *(const v2f*)&lW[(nl + s * 16) * LSTR + k0];
                acc[s] = __builtin_amdgcn_wmma_f32_16x16x4_f32(false, a, false, b,
                                                               (short)0, acc[s],
                                                               false, false);
            }
        }
        __syncthreads();
    }

    const int row0 = bm + wm + ((lane >> 4) << 3);
    const int col0 = bn + (lane & 15);
#pragma unroll
    for (int j = 0; j < 8; ++j) {
        float* gr = &G[(size_t)(row0 + j) * GATES + col0];
#pragma unroll
        for (int s = 0; s < 4; ++s) gr[s * 16] = acc[s][j];
    }
}

// =====================================================================
// LSTM pointwise: gates (i,f,g,o) -> updates c, writes h into AH[:,EMB:]
// =====================================================================
__global__ void lstm_cell_kernel(const float* __restrict__ G,
                                 const float* __restrict__ bih,
                                 const float* __restrict__ bhh,
                                 float* __restrict__ AH, float* __restrict__ C, int B) {
    int t = blockIdx.x * blockDim.x + threadIdx.x;
    if (t >= B * HID) return;
    int m = t / HID, u = t % HID;
    const float* g = &G[(size_t)m * GATES];
    float gi = g[u]           + bih[u]           + bhh[u];
    float gf = g[u + HID]     + bih[u + HID]     + bhh[u + HID];
    float gg = g[u + 2 * HID] + bih[u + 2 * HID] + bhh[u + 2 * HID];
    float go = g[u + 3 * HID] + bih[u + 3 * HID] + bhh[u + 3 * HID];
    float c = sigmoidf_(gf) * C[t] + sigmoidf_(gi) * tanhf(gg);
    C[t] = c;
    AH[m * KAH + EMB + u] = sigmoidf_(go) * tanhf(c);
}

// =====================================================================
// projection GEMM + fused max-pool into out + packed argmax atomics
// logits[B x VTGT] = h @ proj_W^T + b   (K = 512, fp32 WMMA)
// block tile 128x64, wave tile 16x64
// =====================================================================
__global__ __launch_bounds__(256)
void proj_gemm_kernel(const float* __restrict__ H,     // = AH + EMB, lda = KAH
                      const float* __restrict__ W,     // VTGT x HID
                      const float* __restrict__ bias,  // VTGT
                      float* __restrict__ out,         // B x VTGT (max over steps)
                      unsigned long long* __restrict__ packed,
                      int first) {
    __shared__ float lA[TSM * LSTR];
    __shared__ float lW[TSN * LSTR];

    const int tid  = threadIdx.x;
    const int lane = tid & 31;
    const int wv   = tid >> 5;
    const int wm   = wv * 16;
    const int bm   = blockIdx.x * TSM;
    const int bn   = blockIdx.y * TSN;

    const int lrA = tid >> 1;
    const int qA  = (tid & 1) * 8;
    const int lrW = tid >> 2;
    const int qW  = (tid & 3) * 4;

    v8f acc[4] = {v8f{}, v8f{}, v8f{}, v8f{}};

    for (int kb = 0; kb < HID; kb += KC) {
        {
            const float* sp = &H[(bm + lrA) * KAH + kb + qA];
            float4 p0 = ((const float4*)sp)[0];
            float4 p1 = ((const float4*)sp)[1];
            float* d = &lA[lrA * LSTR + qA];
            ((float2*)d)[0]       = make_float2(p0.x, p0.y);
            ((float2*)(d + 2))[0] = make_float2(p0.z, p0.w);
            ((float2*)(d + 4))[0] = make_float2(p1.x, p1.y);
            ((float2*)(d + 6))[0] = make_float2(p1.z, p1.w);
        }
        {
            const float4 p = *(const float4*)&W[(bn + lrW) * HID + kb + qW];
            float* d = &lW[lrW * LSTR + qW];
            ((float2*)d)[0]       = make_float2(p.x, p.y);
            ((float2*)(d + 2))[0] = make_float2(p.z, p.w);
        }
        __syncthreads();

        const int m  = wm + (lane & 15);
        const int nl = lane & 15;
        const int kh = (lane >> 4) * 2;
#pragma unroll
        for (int kk = 0; kk < 4; ++kk) {
            const int k0 = kk * 4 + kh;
            v2f a = *(const v2f*)&lA[m * LSTR + k0];
#pragma unroll
            for (int s = 0; s < 4; ++s) {
                v2f b = *(const v2f*)&lW[(nl + s * 16) * LSTR + k0];
                acc[s] = __builtin_amdgcn_wmma_f32_16x16x4_f32(false, a, false, b,
                                                               (short)0, acc[s],
                                                               false, false);
            }
        }
        __syncthreads();
    }

    const int row0 = bm + wm + ((lane >> 4) << 3);
    const int col0 = bn + (lane & 15);

    unsigned long long pk[8];
#pragma unroll
    for (int j = 0; j < 8; ++j) {
        const int row = row0 + j;
        float* orow = &out[(size_t)row * VTGT + col0];
        unsigned long long best = 0ULL;
#pragma unroll
        for (int s = 0; s < 4; ++s) {
            const int col = col0 + s * 16;
            float l = acc[s][j] + bias[col];
            if (first) orow[s * 16] = l;
            else       orow[s * 16] = fmaxf(orow[s * 16], l);
            unsigned long long p = pack_max(l, col);
            if (p > best) best = p;
        }
        pk[j] = best;
    }
    // butterfly max over the 16 lanes of each half (halves hold disjoint rows)
#pragma unroll
    for (int msk = 8; msk >= 1; msk >>= 1) {
#pragma unroll
        for (int j = 0; j < 8; ++j) {
            unsigned long long o = shfl_xor_u64(pk[j], msk);
            if (o > pk[j]) pk[j] = o;
        }
    }
    if ((lane & 15) == 0) {
#pragma unroll
        for (int j = 0; j < 8; ++j)
            atomicMax(&packed[row0 + j], pk[j]);
    }
}

// =====================================================================
// extract argmax column -> next token; reset packed for next step
// =====================================================================
__global__ void argmax_extract_kernel(unsigned long long* __restrict__ packed,
                                      int* __restrict__ tok, int B) {
    int x = blockIdx.x * blockDim.x + threadIdx.x;
    if (x >= B) return;
    unsigned long long p = packed[x];
    tok[x] = (int)(0xFFFFFFFFu - (unsigned int)(p & 0xFFFFFFFFull));
    packed[x] = 0ULL;
}

// =====================================================================
// launcher
// =====================================================================
extern "C" void kernel_launch(void* const* d_in, const int* in_sizes, int n_in,
                              void* d_out, int out_size, void* d_ws, size_t ws_size,
                              hipStream_t stream) {
    const int*   src     = (const int*)  d_in[0];
    const float* src_emb = (const float*)d_in[1];
    const float* tgt_emb = (const float*)d_in[2];
    const float* eWih    = (const float*)d_in[3];
    const float* eWhh    = (const float*)d_in[4];
    const float* ebih    = (const float*)d_in[5];
    const float* ebhh    = (const float*)d_in[6];
    const float* dWih    = (const float*)d_in[7];
    const float* dWhh    = (const float*)d_in[8];
    const float* dbih    = (const float*)d_in[9];
    const float* dbhh    = (const float*)d_in[10];
    const float* pW      = (const float*)d_in[11];
    const float* pb      = (const float*)d_in[12];
    const int B = in_sizes[0] / 2;   // src is (B, 2)

    // workspace carve-out (256B aligned slices)
    char* ws = (char*)d_ws;
    size_t off = 0;
    auto carve = [&](size_t bytes) -> void* {
        void* p = ws + off;
        off = (off + bytes + 255) & ~(size_t)255;
        return p;
    };
    float* AH  = (float*)carve((size_t)B * KAH   * sizeof(float));  // [x | h]
    float* Cst = (float*)carve((size_t)B * HID   * sizeof(float));  // cell state
    float* G   = (float*)carve((size_t)B * GATES * sizeof(float));  // gate matrix
    int*   tok = (int*)  carve((size_t)B * sizeof(int));
    unsigned long long* packed =
        (unsigned long long*)carve((size_t)B * sizeof(unsigned long long));
    float* out = (float*)d_out;

    const int nInit = B * KAH;
    init_state_kernel<<<(nInit + 255) / 256, 256, 0, stream>>>(AH, Cst, tok, packed, B);

    dim3 gGates(B / TSM, GATES / TSN);   // 16 x 32
    dim3 gProj (B / TSM, VTGT  / TSN);   // 16 x 125
    const int nEmb  = B * EMB;
    const int nCell = B * HID;

    // ---- encoder: 2 steps ----
    for (int t = 0; t < 2; ++t) {
        embed_src_kernel<<<(nEmb + 255) / 256, 256, 0, stream>>>(src, src_emb, AH, t, B);
        gates_gemm_kernel<<<gGates, 256, 0, stream>>>(AH, eWih, eWhh, G);
        lstm_cell_kernel<<<(nCell + 255) / 256, 256, 0, stream>>>(G, ebih, ebhh, AH, Cst, B);
    }
    // ---- greedy decoder: NUM_OUT steps ----
    for (int s = 0; s < NUM_OUT; ++s) {
        embed_tok_kernel<<<(nEmb + 255) / 256, 256, 0, stream>>>(tok, tgt_emb, AH, B);
        gates_gemm_kernel<<<gGates, 256, 0, stream>>>(AH, dWih, dWhh, G);
        lstm_cell_kernel<<<(nCell + 255) / 256, 256, 0, stream>>>(G, dbih, dbhh, AH, Cst, B);
        proj_gemm_kernel<<<gProj, 256, 0, stream>>>(AH + EMB, pW, pb, out, packed,
                                                    s == 0 ? 1 : 0);
        argmax_extract_kernel<<<(B + 255) / 256, 256, 0, stream>>>(packed, tok, B);
    }
}